// VocalEncoder_3264175145347
// MI455X (gfx1250) — compile-verified
//
#include <hip/hip_runtime.h>
#include <hip/hip_bf16.h>

#define N_CLASS 384
#define D_MODEL 128
#define W_COLS  (N_CLASS + 1)   // 385

typedef float v4f __attribute__((ext_vector_type(4)));

// ---------------------------------------------------------------------------
// Kernel A: build fused, transposed embedding table in workspace.
//   table[c*128 + d] = weight[d*385 + c] + bias[d]   (c < 384)
//   gcol[d]          = weight[d*385 + 384]
// One-shot, ~50K threads; cost is noise vs. the 256 MB main stream.
// ---------------------------------------------------------------------------
__global__ void __launch_bounds__(256) vocal_build_table(
    const float* __restrict__ w, const float* __restrict__ bias,
    float* __restrict__ table, float* __restrict__ gcol)
{
    int i = blockIdx.x * 256 + threadIdx.x;
    if (i < N_CLASS * D_MODEL) {
        int c = i >> 7;       // i / 128
        int d = i & 127;      // i % 128
        table[i] = w[(long)d * W_COLS + c] + bias[d];
    } else if (i < N_CLASS * D_MODEL + D_MODEL) {
        int d = i - N_CLASS * D_MODEL;
        gcol[d] = w[(long)d * W_COLS + N_CLASS];
    }
}

// ---------------------------------------------------------------------------
// Kernel B: main gather. One wave32 per 32 tokens; lane L owns dims 4L..4L+3.
//  - pitch/gain staged via GLOBAL_LOAD_ASYNC_TO_LDS_B32 (+ s_wait_asynccnt)
//  - table row gather = one global_load_b128 per lane (L2-resident table)
//  - output via non-temporal b128 stores (don't evict the table from L2)
// Grid is sized so every block is full: EXEC is all-ones for the async ops.
// ---------------------------------------------------------------------------
__global__ void __launch_bounds__(256) vocal_gather(
    const int*   __restrict__ pitch,
    const float* __restrict__ gain,
    const float* __restrict__ table,
    const float* __restrict__ gcol,
    float*       __restrict__ out)
{
    __shared__ int   s_pitch[256];
    __shared__ float s_gain[256];

    const int  lane  = threadIdx.x & 31;
    // first token of this wave (32 tokens per wave, 8 waves per block)
    const long tok0  = (long)blockIdx.x * 256 + (threadIdx.x & ~31);

    // ---- async stage this wave's 32 pitches + 32 gains into LDS ----
    unsigned ldsP = (unsigned)(unsigned long long)&s_pitch[threadIdx.x];
    unsigned ldsG = (unsigned)(unsigned long long)&s_gain[threadIdx.x];
    unsigned long long gP = (unsigned long long)(const void*)(pitch + tok0 + lane);
    unsigned long long gG = (unsigned long long)(const void*)(gain  + tok0 + lane);
    asm volatile("global_load_async_to_lds_b32 %0, %1, off"
                 :: "v"(ldsP), "v"(gP) : "memory");
    asm volatile("global_load_async_to_lds_b32 %0, %1, off"
                 :: "v"(ldsG), "v"(gG) : "memory");

    // per-lane slice of the gain column (amortized over 32 tokens)
    v4f gc = *(const v4f*)(gcol + 4 * lane);

    asm volatile("s_wait_asynccnt 0x0" ::: "memory");

    int   c = s_pitch[threadIdx.x];
    float g = s_gain[threadIdx.x];

    float* outw = out + tok0 * D_MODEL + lane * 4;

    #pragma unroll
    for (int i = 0; i < 32; ++i) {
        int   ci = __builtin_amdgcn_readlane(c, i);
        float gi = __int_as_float(__builtin_amdgcn_readlane(__float_as_int(g), i));
        // contiguous 512B row, lane L reads 16B at offset 16L -> global_load_b128
        v4f e = *((const v4f*)(table + (long)ci * D_MODEL) + lane);
        v4f o;
        o.x = fmaf(gi, gc.x, e.x);
        o.y = fmaf(gi, gc.y, e.y);
        o.z = fmaf(gi, gc.z, e.z);
        o.w = fmaf(gi, gc.w, e.w);
        __builtin_nontemporal_store(o, (v4f*)(outw + (long)i * D_MODEL));
    }
}

// ---------------------------------------------------------------------------
// Fallback (workspace too small or tokens % 256 != 0): strided gather, correct but slow.
// ---------------------------------------------------------------------------
__global__ void __launch_bounds__(256) vocal_fallback(
    const int* __restrict__ pitch, const float* __restrict__ gain,
    const float* __restrict__ w, const float* __restrict__ bias,
    float* __restrict__ out, long n)
{
    long i = (long)blockIdx.x * 256 + threadIdx.x;
    if (i >= n) return;
    int  d = (int)(i & (D_MODEL - 1));
    long t = i >> 7;
    int  c = pitch[t];
    out[i] = w[(long)d * W_COLS + c] + gain[t] * w[(long)d * W_COLS + N_CLASS] + bias[d];
}

extern "C" void kernel_launch(void* const* d_in, const int* in_sizes, int n_in,
                              void* d_out, int out_size, void* d_ws, size_t ws_size,
                              hipStream_t stream) {
    const int*   pitch  = (const int*)  d_in[0];
    const float* gain   = (const float*)d_in[1];
    const float* weight = (const float*)d_in[2];
    const float* bias   = (const float*)d_in[3];
    float*       out    = (float*)d_out;

    const long tokens = in_sizes[0];                  // 64*8192 = 524288
    const size_t needWs = (size_t)(N_CLASS * D_MODEL + D_MODEL) * sizeof(float);

    if (ws_size >= needWs && (tokens % 256) == 0) {
        float* table = (float*)d_ws;
        float* gcol  = table + N_CLASS * D_MODEL;
        int blocksA = (N_CLASS * D_MODEL + D_MODEL + 255) / 256;
        vocal_build_table<<<blocksA, 256, 0, stream>>>(weight, bias, table, gcol);
        int blocksB = (int)(tokens / 256);            // 256 tokens per block
        vocal_gather<<<blocksB, 256, 0, stream>>>(pitch, gain, table, gcol, out);
    } else {
        long n = tokens * D_MODEL;
        int blocks = (int)((n + 255) / 256);
        vocal_fallback<<<blocks, 256, 0, stream>>>(pitch, gain, weight, bias, out, n);
    }
}